// Voxelization_18614388260890
// MI455X (gfx1250) — compile-verified
//
#include <hip/hip_runtime.h>

// ---------------- problem constants (must match reference) ----------------
#define N_PTS    2000000
#define GRID_X   432
#define GRID_Y   496
#define GRID_Z   1
#define G_TOTAL  (GRID_X * GRID_Y * GRID_Z)     // 214272
#define MAX_PTS  32
#define MAX_VOX  20000
#define SENT     0x7FFFFFFF

#define SCAN_CHUNK 1024
#define NBLK ((N_PTS + SCAN_CHUNK - 1) / SCAN_CHUNK)   // 1954

// ---------------- workspace layout (int32 element offsets) ----------------
#define OFF_LIN   0
#define OFF_FIRST (OFF_LIN   + N_PTS)
#define OFF_CRANK (OFF_FIRST + G_TOTAL + 1)
#define OFF_RCELL (OFF_CRANK + G_TOTAL + 1)
#define OFF_BSUM  (OFF_RCELL + MAX_VOX)
#define OFF_BOFF  (OFF_BSUM  + NBLK)
#define OFF_NACT  (OFF_BOFF  + NBLK)
#define OFF_SLOT  (OFF_NACT  + 1)               // MAX_VOX*MAX_PTS ints

// ---------------- output layout (float element offsets) -------------------
#define OUT_VOX   0
#define OUT_COORS (MAX_VOX * MAX_PTS * 4)       // 2,560,000
#define OUT_NP    (OUT_COORS + MAX_VOX * 3)     // 2,620,000
#define OUT_NV    (OUT_NP + MAX_VOX)            // 2,640,000

// Voxel coordinate -> linear cell id (G_TOTAL if out of range).
// Must match JAX: floor((p - lo)/vs) in IEEE f32.
__device__ __forceinline__ int lin_of(float x, float y, float z) {
    int cx = (int)floorf((x - 0.0f)   / 0.16f);
    int cy = (int)floorf((y + 39.68f) / 0.16f);
    int cz = (int)floorf((z + 3.0f)   / 4.0f);
    bool ok = (cx >= 0) & (cx < GRID_X) &
              (cy >= 0) & (cy < GRID_Y) &
              (cz >= 0) & (cz < GRID_Z);
    return ok ? (cz * GRID_Y + cy) * GRID_X + cx : G_TOTAL;
}

// -------- init: first=N, cellRank=SENT, slots=SENT (fresh every call) -----
__global__ void k_init(int* __restrict__ first, int* __restrict__ crank,
                       int* __restrict__ slot) {
    int i = blockIdx.x * 256 + threadIdx.x;
    if (i < MAX_VOX * MAX_PTS) slot[i] = SENT;
    if (i <= G_TOTAL) { first[i] = N_PTS; crank[i] = SENT; }
}

// -------- pass 1: per-point cell id + first-point-of-cell via atomicMin ----
__global__ void k_first(const float4* __restrict__ pts, int* __restrict__ lin,
                        int* __restrict__ first) {
    int p = blockIdx.x * 256 + threadIdx.x;
    if (p >= N_PTS) return;
    float4 pt = pts[p];                          // global_load_b128
    int l = lin_of(pt.x, pt.y, pt.z);
    lin[p] = l;
    if (l != G_TOTAL) atomicMin(&first[l], p);   // order-independent => deterministic
}

// -------- pass 2a: per-1024-point block, count first-of-cell flags ---------
__global__ void k_blocksum(const int* __restrict__ lin,
                           const int* __restrict__ first,
                           int* __restrict__ bsum) {
    __shared__ int sh[256];
    int t = threadIdx.x;
    int base = blockIdx.x * SCAN_CHUNK + t * 4;
    int tot = 0;
    if (base < N_PTS) {                          // N_PTS % 4 == 0 => all 4 valid
        int4 l4 = *(const int4*)(lin + base);    // b128 load
        if (l4.x != G_TOTAL && first[l4.x] == base + 0) tot++;
        if (l4.y != G_TOTAL && first[l4.y] == base + 1) tot++;
        if (l4.z != G_TOTAL && first[l4.z] == base + 2) tot++;
        if (l4.w != G_TOTAL && first[l4.w] == base + 3) tot++;
    }
    sh[t] = tot; __syncthreads();
    for (int off = 128; off > 0; off >>= 1) {
        if (t < off) sh[t] += sh[t + off];
        __syncthreads();
    }
    if (t == 0) bsum[blockIdx.x] = sh[0];
}

// -------- pass 2b: single-block exclusive scan of the 1954 block sums ------
__global__ void k_scan(const int* __restrict__ bsum, int* __restrict__ boff,
                       int* __restrict__ nact, float* __restrict__ out) {
    __shared__ int sh[256];
    __shared__ int carry;
    int t = threadIdx.x;
    if (t == 0) carry = 0;
    __syncthreads();
    for (int base = 0; base < NBLK; base += 256) {
        int i = base + t;
        int v = (i < NBLK) ? bsum[i] : 0;
        sh[t] = v; __syncthreads();
        for (int off = 1; off < 256; off <<= 1) {
            int u = (t >= off) ? sh[t - off] : 0;
            __syncthreads();
            sh[t] += u;
            __syncthreads();
        }
        if (i < NBLK) boff[i] = carry + sh[t] - v;   // exclusive + carry
        int tileTot = sh[255];
        __syncthreads();
        if (t == 0) carry += tileTot;
        __syncthreads();
    }
    if (t == 0) {
        nact[0] = carry;                              // #active voxels
        int nv = carry < MAX_VOX ? carry : MAX_VOX;
        out[OUT_NV] = (float)nv;                      // num_voxels
    }
}

// -------- pass 2c: assign first-appearance ranks to cells ------------------
__global__ void k_rank(const int* __restrict__ lin,
                       const int* __restrict__ first,
                       const int* __restrict__ boff,
                       int* __restrict__ crank, int* __restrict__ rcell) {
    __shared__ int sh[256];
    int t = threadIdx.x;
    int base = blockIdx.x * SCAN_CHUNK + t * 4;
    int l[4] = {G_TOTAL, G_TOTAL, G_TOTAL, G_TOTAL};
    int f[4] = {0, 0, 0, 0};
    int tot = 0;
    if (base < N_PTS) {
        int4 l4 = *(const int4*)(lin + base);
        l[0] = l4.x; l[1] = l4.y; l[2] = l4.z; l[3] = l4.w;
        #pragma unroll
        for (int k = 0; k < 4; k++) {
            f[k] = (l[k] != G_TOTAL && first[l[k]] == base + k) ? 1 : 0;
            tot += f[k];
        }
    }
    sh[t] = tot; __syncthreads();
    for (int off = 1; off < 256; off <<= 1) {
        int u = (t >= off) ? sh[t - off] : 0;
        __syncthreads();
        sh[t] += u;
        __syncthreads();
    }
    int r = boff[blockIdx.x] + sh[t] - tot;           // rank of first flagged pt
    #pragma unroll
    for (int k = 0; k < 4; k++) {
        if (f[k]) {
            if (r < MAX_VOX) { crank[l[k]] = r; rcell[r] = l[k]; }
            r++;
        }
    }
}

// -------- pass 3: lock-free sorted insertion of the 32 smallest indices ----
// atomicMin cascade: final slot state == sorted 32 smallest point indices,
// independent of thread interleaving => deterministic and matches the
// reference's "slot = rank by point index, keep first 32".
__global__ void k_insert(const int* __restrict__ lin,
                         const int* __restrict__ crank,
                         int* __restrict__ slot) {
    int p = blockIdx.x * 256 + threadIdx.x;
    if (p >= N_PTS) return;
    int l = lin[p];
    if (l == G_TOTAL) return;
    int r = crank[l];
    if (r >= MAX_VOX) return;                         // SENT covers inactive/overflow ranks
    int cur = p;
    int b = r * MAX_PTS;
    #pragma unroll 4
    for (int s = 0; s < MAX_PTS; s++) {
        int prev = atomicMin(&slot[b + s], cur);
        if (prev == SENT) break;                      // landed in empty slot
        cur = (prev > cur) ? prev : cur;              // carry the displaced max
    }
}

// -------- pass 4: one wave32 per voxel; b128 gather + b128 store -----------
// Slots are sorted ascending with SENT at the tail, so the point count is
// popc(ballot(valid)) — wave32-native, no LDS, no shared atomics.
__global__ void k_out(const float4* __restrict__ pts4,
                      const int* __restrict__ slot,
                      const int* __restrict__ rcell,
                      const int* __restrict__ nact,
                      float* __restrict__ out) {
    int t    = threadIdx.x;
    int r    = blockIdx.x * 8 + (t >> 5);             // 8 voxels per 256-thr block
    int lane = t & 31;

    int p = slot[r * MAX_PTS + lane];                 // coalesced b32
    bool valid = (p != SENT);
    unsigned mask = __builtin_amdgcn_ballot_w32(valid);

    float4 v = make_float4(0.0f, 0.0f, 0.0f, 0.0f);
    if (valid) v = pts4[p];                           // b128 gather (L2-resident)
    ((float4*)(out + OUT_VOX))[r * MAX_PTS + lane] = v;  // b128 store

    if (lane == 0) out[OUT_NP + r] = (float)__popc(mask);
    if (lane < 3) {
        float cv = -1.0f;
        if (r < nact[0]) {
            int cell = rcell[r];
            int cz = cell / (GRID_X * GRID_Y);
            int rem = cell % (GRID_X * GRID_Y);
            int cy = rem / GRID_X;
            int cx = rem % GRID_X;
            cv = (float)((lane == 0) ? cz : ((lane == 1) ? cy : cx));
        }
        out[OUT_COORS + r * 3 + lane] = cv;
    }
}

extern "C" void kernel_launch(void* const* d_in, const int* in_sizes, int n_in,
                              void* d_out, int out_size, void* d_ws, size_t ws_size,
                              hipStream_t stream) {
    const float* pts = (const float*)d_in[0];
    float* out = (float*)d_out;
    int* ws = (int*)d_ws;

    int* lin   = ws + OFF_LIN;
    int* first = ws + OFF_FIRST;
    int* crank = ws + OFF_CRANK;
    int* rcell = ws + OFF_RCELL;
    int* bsum  = ws + OFF_BSUM;
    int* boff  = ws + OFF_BOFF;
    int* nact  = ws + OFF_NACT;
    int* slot  = ws + OFF_SLOT;

    const int ptBlocks   = (N_PTS + 255) / 256;                 // 7813
    const int initBlocks = (MAX_VOX * MAX_PTS + 255) / 256;     // 2500 (covers G_TOTAL+1 too)

    k_init    <<<initBlocks, 256, 0, stream>>>(first, crank, slot);
    k_first   <<<ptBlocks,   256, 0, stream>>>((const float4*)pts, lin, first);
    k_blocksum<<<NBLK,       256, 0, stream>>>(lin, first, bsum);
    k_scan    <<<1,          256, 0, stream>>>(bsum, boff, nact, out);
    k_rank    <<<NBLK,       256, 0, stream>>>(lin, first, boff, crank, rcell);
    k_insert  <<<ptBlocks,   256, 0, stream>>>(lin, crank, slot);
    k_out     <<<MAX_VOX/8,  256, 0, stream>>>((const float4*)pts, slot, rcell, nact, out);
}